// GCN_19928648253615
// MI455X (gfx1250) — compile-verified
//
#include <hip/hip_runtime.h>
#include <math.h>

typedef float v2f __attribute__((ext_vector_type(2)));
typedef float v8f __attribute__((ext_vector_type(8)));

#define NFEAT 256
#define NHID  128
#define NCLASS 8

// ---------------------------------------------------------------------------
// init: out2[i][f] = b1[f]  (bias folded into the aggregation buffer),
//       deg[i] = 1          (self-loop contribution to degree)
// ---------------------------------------------------------------------------
__global__ void k_init(float* __restrict__ out2, unsigned* __restrict__ deg,
                       const float* __restrict__ b1, int nNodes) {
    int tid = blockIdx.x * blockDim.x + threadIdx.x;
    int total = nNodes * NHID;
    if (tid < total) out2[tid] = b1[tid & (NHID - 1)];
    if (tid < nNodes) deg[tid] = 1u;
}

// ---------------------------------------------------------------------------
// degree: deg[dst[e]] += 1 over real edges (int64 edge indices)
// ---------------------------------------------------------------------------
__global__ void k_degree(const long long* __restrict__ ei,
                         unsigned* __restrict__ deg, int nEdges) {
    int e = blockIdx.x * blockDim.x + threadIdx.x;
    if (e < nEdges) {
        int d = (int)ei[(size_t)nEdges + e];
        __hip_atomic_fetch_add(&deg[d], 1u, __ATOMIC_RELAXED,
                               __HIP_MEMORY_SCOPE_AGENT);
    }
}

// ---------------------------------------------------------------------------
// dinv[i] = deg[i] > 0 ? rsqrt(deg[i]) : 0
// ---------------------------------------------------------------------------
__global__ void k_dinv(const unsigned* __restrict__ deg,
                       float* __restrict__ dinv, int nNodes) {
    int i = blockIdx.x * blockDim.x + threadIdx.x;
    if (i < nNodes) {
        float d = (float)deg[i];
        dinv[i] = d > 0.f ? rsqrtf(d) : 0.f;
    }
}

// ---------------------------------------------------------------------------
// H = X @ W1 via V_WMMA_F32_16X16X4_F32 (fp32, matches reference precision).
// Block = 256 threads = 8 waves; each wave owns a private 16-row tile
// (128 rows per block) and sweeps all 8 N-tiles of NHID=128.
// A frag: lane<16 -> K={k0,k0+1}, lane>=16 -> K={k0+2,k0+3}, row = lane&15.
// B frag: VGPR0 = W1 rows k0 / k0+2 (lane halves), VGPR1 = rows k0+1 / k0+3.
// ---------------------------------------------------------------------------
__global__ void k_gemm1(const float* __restrict__ X, const float* __restrict__ W1,
                        float* __restrict__ H, int nNodes) {
    const int lane    = threadIdx.x & 31;
    const int wave    = threadIdx.x >> 5;
    const int rowBase = blockIdx.x * 128 + wave * 16;
    if (rowBase >= nNodes) return;                 // wave-uniform exit

    const int mrow = rowBase + (lane & 15);
    const int arow = mrow < nNodes ? mrow : nNodes - 1;   // clamp, EXEC stays full
    const int koff = (lane >> 4) * 2;              // 0 or 2
    const int bcol = lane & 15;
    const int mOff = (lane >> 4) * 8;              // C/D half-split

    for (int nt = 0; nt < NHID / 16; ++nt) {
        const int n0 = nt * 16;
        v8f c = {};
        const float* ap = X  + (size_t)arow * NFEAT + koff;
        const float* bp = W1 + (size_t)koff * NHID + n0 + bcol;
#pragma unroll 8
        for (int k = 0; k < NFEAT; k += 4) {
            v2f a = *(const v2f*)ap;               // 8B-aligned contiguous pair
            v2f b; b.x = bp[0]; b.y = bp[NHID];
            c = __builtin_amdgcn_wmma_f32_16x16x4_f32(
                    false, a, false, b, (short)0, c, false, false);
            ap += 4;
            bp += 4 * NHID;
        }
#pragma unroll
        for (int r = 0; r < 8; ++r) {
            int m = rowBase + mOff + r;
            if (m < nNodes) H[(size_t)m * NHID + n0 + bcol] = c[r];
        }
    }
}

// ---------------------------------------------------------------------------
// Scatter: one wave32 per unit (edge, or self-loop for unit >= nEdges).
// Each lane owns a float4 slice of the 128-wide feature row; accumulation via
// hardware global_atomic_add_f32 (relaxed, agent scope) into L2-resident out2.
// ---------------------------------------------------------------------------
__global__ void k_scatter(const long long* __restrict__ ei,
                          const float* __restrict__ h,
                          const float* __restrict__ dinv,
                          float* __restrict__ out2,
                          int nEdges, int nNodes) {
    const int lane = threadIdx.x & 31;
    const int unit = blockIdx.x * (blockDim.x >> 5) + (threadIdx.x >> 5);
    if (unit >= nEdges + nNodes) return;

    int s, d;
    if (unit < nEdges) {
        s = (int)ei[unit];
        d = (int)ei[(size_t)nEdges + unit];
    } else {
        s = d = unit - nEdges;                     // self-loop
    }
    const float w = dinv[s] * dinv[d];
    const float4 hv = *(const float4*)(h + (size_t)s * NHID + lane * 4);
    float* o = out2 + (size_t)d * NHID + lane * 4;
    __hip_atomic_fetch_add(o + 0, hv.x * w, __ATOMIC_RELAXED, __HIP_MEMORY_SCOPE_AGENT);
    __hip_atomic_fetch_add(o + 1, hv.y * w, __ATOMIC_RELAXED, __HIP_MEMORY_SCOPE_AGENT);
    __hip_atomic_fetch_add(o + 2, hv.z * w, __ATOMIC_RELAXED, __HIP_MEMORY_SCOPE_AGENT);
    __hip_atomic_fetch_add(o + 3, hv.w * w, __ATOMIC_RELAXED, __HIP_MEMORY_SCOPE_AGENT);
}

// ---------------------------------------------------------------------------
// out = out2 @ Wfc + bfc via fp32 WMMA; K=128, single N-tile with columns
// 8..15 zero-padded (NCLASS=8). Same per-wave 16-row tiling as k_gemm1.
// ---------------------------------------------------------------------------
__global__ void k_fc(const float* __restrict__ H2, const float* __restrict__ Wfc,
                     const float* __restrict__ bfc, float* __restrict__ out,
                     int nNodes) {
    const int lane    = threadIdx.x & 31;
    const int wave    = threadIdx.x >> 5;
    const int rowBase = blockIdx.x * 128 + wave * 16;
    if (rowBase >= nNodes) return;

    const int mrow = rowBase + (lane & 15);
    const int arow = mrow < nNodes ? mrow : nNodes - 1;
    const int koff = (lane >> 4) * 2;
    const int bcol = lane & 15;
    const int ccol = bcol < NCLASS ? bcol : NCLASS - 1;   // safe pointer base
    const bool colOk = bcol < NCLASS;

    v8f c = {};
    const float* ap = H2  + (size_t)arow * NHID + koff;
    const float* bp = Wfc + (size_t)koff * NCLASS + ccol;
#pragma unroll 8
    for (int k = 0; k < NHID; k += 4) {
        v2f a = *(const v2f*)ap;
        v2f b;
        b.x = colOk ? bp[0]      : 0.f;
        b.y = colOk ? bp[NCLASS] : 0.f;
        c = __builtin_amdgcn_wmma_f32_16x16x4_f32(
                false, a, false, b, (short)0, c, false, false);
        ap += 4;
        bp += 4 * NCLASS;
    }
    if (colOk) {
        const float bias = bfc[bcol];
        const int mOff = (lane >> 4) * 8;
#pragma unroll
        for (int r = 0; r < 8; ++r) {
            int m = rowBase + mOff + r;
            if (m < nNodes) out[(size_t)m * NCLASS + bcol] = c[r] + bias;
        }
    }
}

// ---------------------------------------------------------------------------
extern "C" void kernel_launch(void* const* d_in, const int* in_sizes, int n_in,
                              void* d_out, int out_size, void* d_ws, size_t ws_size,
                              hipStream_t stream) {
    const float*     X   = (const float*)d_in[0];
    const long long* ei  = (const long long*)d_in[1];   // int64 edge_index [2, E]
    const float*     W1  = (const float*)d_in[2];
    const float*     b1  = (const float*)d_in[3];
    const float*     Wfc = (const float*)d_in[4];
    const float*     bfc = (const float*)d_in[5];
    float*           out = (float*)d_out;

    const int nNodes = in_sizes[0] / NFEAT;
    const int nEdges = in_sizes[1] / 2;

    // Workspace carve-out (all re-initialized every launch; deterministic).
    char*  ws  = (char*)d_ws;
    size_t off = 0;
    auto carve = [&](size_t bytes) {
        void* p = ws + off;
        off = (off + bytes + 255) & ~(size_t)255;
        return p;
    };
    float*    H    = (float*)carve((size_t)nNodes * NHID * sizeof(float));
    float*    OUT2 = (float*)carve((size_t)nNodes * NHID * sizeof(float));
    unsigned* DEG  = (unsigned*)carve((size_t)nNodes * sizeof(unsigned));
    float*    DINV = (float*)carve((size_t)nNodes * sizeof(float));
    (void)ws_size; (void)n_in; (void)out_size;

    const int totInit = nNodes * NHID;
    k_init  <<<(totInit + 255) / 256, 256, 0, stream>>>(OUT2, DEG, b1, nNodes);
    k_degree<<<(nEdges  + 255) / 256, 256, 0, stream>>>(ei, DEG, nEdges);
    k_dinv  <<<(nNodes  + 255) / 256, 256, 0, stream>>>(DEG, DINV, nNodes);
    k_gemm1 <<<(nNodes  + 127) / 128, 256, 0, stream>>>(X, W1, H, nNodes);

    const int units = nEdges + nNodes;                 // edges + self-loops
    k_scatter<<<(units + 7) / 8, 256, 0, stream>>>(ei, H, DINV, OUT2, nEdges, nNodes);

    k_fc    <<<(nNodes + 127) / 128, 256, 0, stream>>>(OUT2, Wfc, bfc, out, nNodes);
}